// mPD_loss_2_54941221651104
// MI455X (gfx1250) — compile-verified
//
#include <hip/hip_runtime.h>
#include <math.h>

typedef __attribute__((ext_vector_type(2))) float v2f;
typedef __attribute__((ext_vector_type(8))) float v8f;

#define BATCH 2048
#define NPTS  4095
#define NPOS  4096                   // N+1 scan length
#define ROWPAD 18                    // 16 cols padded to 18 floats (8B-aligned pairs, conflict-free)
#define CHUNK_FLOATS (16 * ROWPAD)   // one 16x16 tile = 288 floats
#define COORD_FLOATS (16 * CHUNK_FLOATS) // 16 chunks per coordinate = 4608 floats
#define NCHUNK 16

// Phase 1: coalesced global loads + trig once per point -> D values staged in LDS.
// Phase 2: per-coordinate wave does 16x (16x16 tile scan via chained v_wmma_f32_16x16x4_f32
//          with an upper-triangular ones B matrix), carries via v_readlane, |.| accumulate.
__global__ __launch_bounds__(256) void mpd_main_kernel(
    const float* __restrict__ o3, const float* __restrict__ s3,
    const float* __restrict__ o2, const float* __restrict__ s2,
    const float* __restrict__ df, float* __restrict__ partial)
{
    __shared__ float lds[3 * COORD_FLOATS];   // 55,296 bytes

    const int b = blockIdx.x;
    const int t = threadIdx.x;

    const float* s3b = s3 + (size_t)b * 3 * NPTS;
    const float* s2b = s2 + (size_t)b * 3 * NPTS;
    const float* dfb = df + (size_t)b * 2 * NPTS;

    // element e=0 of each coordinate scan: origin difference
    if (t < 3) {
        lds[t * COORD_FLOATS] = o3[b * 3 + t] - o2[b * 3 + t];
    }

    // elements e=1..4095  (point j -> position e=j+1)
    for (int j = t; j < NPTS; j += 256) {
        float r3  = s3b[j];
        float th3 = s3b[NPTS + j]     + dfb[j];
        float ph3 = s3b[2 * NPTS + j] + dfb[NPTS + j];
        float r2  = s2b[j];
        float th2 = s2b[NPTS + j];
        float ph2 = s2b[2 * NPTS + j];

        float st3 = __sinf(th3), ct3 = __cosf(th3);
        float sp3 = __sinf(ph3), cp3 = __cosf(ph3);
        float st2 = __sinf(th2), ct2 = __cosf(th2);
        float sp2 = __sinf(ph2), cp2 = __cosf(ph2);

        float dx = r3 * st3 * cp3 - r2 * st2 * cp2;
        float dy = r3 * st3 * sp3 - r2 * st2 * sp2;
        float dz = r3 * ct3       - r2 * ct2;

        int e   = j + 1;
        int idx = ((e >> 8) * CHUNK_FLOATS) + (((e >> 4) & 15) * ROWPAD) + (e & 15);
        lds[idx]                    = dx;
        lds[COORD_FLOATS + idx]     = dy;
        lds[2 * COORD_FLOATS + idx] = dz;
    }
    __syncthreads();

    const int w = t >> 5;           // wave id; waves 0..2 scan coordinates x/y/z
    if (w < 3) {
        const int lane = t & 31;
        const int row  = lane & 15; // A-matrix M index for this lane
        const int hi   = lane >> 4; // lanes 16-31 carry K=2,3 halves
        const float* cb = lds + w * COORD_FLOATS;

        float base = 0.0f;          // uniform inter-chunk carry
        float acc  = 0.0f;          // per-lane sum of |prefix|

        for (int k = 0; k < NCHUNK; ++k) {
            const float* chb = cb + k * CHUNK_FLOATS + row * ROWPAD + 2 * hi;

            v8f C = {};
            #pragma unroll
            for (int p = 0; p < 4; ++p) {
                // A operand (16x4 slice of X): lane L<16 -> X[row,4p],X[row,4p+1];
                // lanes 16-31 -> X[row,4p+2],X[row,4p+3]. 8B-aligned ds pair load.
                float2 av = *reinterpret_cast<const float2*>(chb + 4 * p);
                v2f A; A.x = av.x; A.y = av.y;
                // B operand: rows 4p..4p+3 of upper-triangular ones U (U[k,n]=1 iff k<=n)
                int k0 = 4 * p + 2 * hi;
                v2f Bv;
                Bv.x = (row >= k0)     ? 1.0f : 0.0f;
                Bv.y = (row >= k0 + 1) ? 1.0f : 0.0f;
                C = __builtin_amdgcn_wmma_f32_16x16x4_f32(
                        false, A, false, Bv, (short)0, C, false, false);
            }

            // row totals live in column 15: lane 15 (rows 0-7), lane 31 (rows 8-15)
            float tot[16];
            #pragma unroll
            for (int r = 0; r < 8; ++r) {
                tot[r]     = __int_as_float(__builtin_amdgcn_readlane(__float_as_int(C[r]), 15));
                tot[r + 8] = __int_as_float(__builtin_amdgcn_readlane(__float_as_int(C[r]), 31));
            }
            float carry[16];
            carry[0] = 0.0f;
            #pragma unroll
            for (int i = 1; i < 16; ++i) carry[i] = carry[i - 1] + tot[i - 1];
            float chunkTotal = carry[15] + tot[15];

            #pragma unroll
            for (int r = 0; r < 8; ++r) {
                float add = base + (hi ? carry[r + 8] : carry[r]);
                acc += fabsf(C[r] + add);
            }
            base += chunkTotal;
        }

        // wave32 reduction of acc
        #pragma unroll
        for (int m = 16; m >= 1; m >>= 1) acc += __shfl_xor(acc, m, 32);
        if (lane == 0) partial[b * 3 + w] = acc;
    }
}

// Deterministic final reduction: sum 2048*3 partials, scale by 1/4096 (mean over positions).
__global__ __launch_bounds__(256) void mpd_reduce_kernel(
    const float* __restrict__ partial, float* __restrict__ out)
{
    __shared__ float sm[256];
    float s = 0.0f;
    for (int i = threadIdx.x; i < BATCH * 3; i += 256) s += partial[i];
    sm[threadIdx.x] = s;
    __syncthreads();
    for (int n = 128; n >= 1; n >>= 1) {
        if ((int)threadIdx.x < n) sm[threadIdx.x] += sm[threadIdx.x + n];
        __syncthreads();
    }
    if (threadIdx.x == 0) out[0] = sm[0] * (1.0f / 4096.0f);
}

extern "C" void kernel_launch(void* const* d_in, const int* in_sizes, int n_in,
                              void* d_out, int out_size, void* d_ws, size_t ws_size,
                              hipStream_t stream) {
    const float* o3 = (const float*)d_in[0];   // origin_3D        (B,3,1)
    const float* s3 = (const float*)d_in[1];   // spherical_3D     (B,3,N)
    const float* o2 = (const float*)d_in[2];   // origin_2D        (B,3,1)
    const float* s2 = (const float*)d_in[3];   // spherical_2D     (B,3,N)
    const float* df = (const float*)d_in[4];   // deformation_field(B,2,N)
    float* out      = (float*)d_out;
    float* partial  = (float*)d_ws;            // BATCH*3 floats = 24 KB scratch

    mpd_main_kernel<<<BATCH, 256, 0, stream>>>(o3, s3, o2, s2, df, partial);
    mpd_reduce_kernel<<<1, 256, 0, stream>>>(partial, out);
}